// Attention_35785667510463
// MI455X (gfx1250) — compile-verified
//
#include <hip/hip_runtime.h>
#include <hip/hip_bf16.h>
#include <math.h>

#define NEG_INF_F (-1.0e10f)

typedef __attribute__((ext_vector_type(2))) float v2f;
typedef __attribute__((ext_vector_type(8))) float v8f;

// ---------------------------------------------------------------------------
// Kernel 1: u[c] = sum_a v[a] * W[a, c]   (u = W^T v, c in [0, 2048))
//           also c0 = dot(v, b_attn)
// W is [1024, 2048] row-major -> reads coalesced across c.
// ---------------------------------------------------------------------------
__global__ void k_project_v(const float* __restrict__ W,
                            const float* __restrict__ v,
                            const float* __restrict__ b_attn,
                            float* __restrict__ u,
                            float* __restrict__ c0) {
    int c = blockIdx.x * blockDim.x + threadIdx.x;   // 0..2047
    float acc = 0.f;
    for (int a = 0; a < 1024; ++a)
        acc = fmaf(v[a], W[(size_t)a * 2048 + c], acc);
    u[c] = acc;
    if (blockIdx.x == 0 && threadIdx.x == 0) {
        float s = 0.f;
        for (int a = 0; a < 1024; ++a) s = fmaf(v[a], b_attn[a], s);
        *c0 = s;
    }
}

// ---------------------------------------------------------------------------
// Kernel 2: score_h[b] = c0 + dot(attention_vector[b, :], u[0:1024])
// One block per b, 256-thread tree reduction.
// ---------------------------------------------------------------------------
__global__ void k_score_h(const float* __restrict__ av,
                          const float* __restrict__ u,
                          const float* __restrict__ c0,
                          float* __restrict__ score_h) {
    __shared__ float red[256];
    int b = blockIdx.x;
    float acc = 0.f;
    for (int d = threadIdx.x; d < 1024; d += 256)
        acc = fmaf(av[(size_t)b * 1024 + d], u[d], acc);
    red[threadIdx.x] = acc;
    __syncthreads();
    for (int s = 128; s > 0; s >>= 1) {
        if (threadIdx.x < (unsigned)s) red[threadIdx.x] += red[threadIdx.x + s];
        __syncthreads();
    }
    if (threadIdx.x == 0) score_h[b] = red[0] + *c0;
}

// ---------------------------------------------------------------------------
// Kernel 3 (dominant, HBM-bound): WMMA batched dot over encoder_outputs.
//   scores[b, t] = mask ? (score_h[b] + u_e . enc[t, b, :]) / log2(|t-ts|+2)
//                       : NEG_INF
// One wave per (t, 16-consecutive-b) tile. A = 16x4 fp32 tile of enc rows
// (ISA layout: lanes 0-15 hold K=0..1, lanes 16-31 hold K=2..3 -> one float2
// per lane). B = u_e chunk broadcast across all 16 columns, so every column
// of D holds the score (independent of B's N mapping).
// D layout (C/D 16x16 f32): VGPR j = row j (lanes 0-15) and row j+8 (16-31).
// ---------------------------------------------------------------------------
__global__ void k_score_wmma(const float* __restrict__ enc,     // [T=1024, B=64, E2=1024]
                             const float* __restrict__ u_e,     // [1024]
                             const float* __restrict__ score_h, // [64]
                             const int*   __restrict__ mask,    // [64, 1024]
                             const int*   __restrict__ time_step_p,
                             float* __restrict__ scores) {      // [64, 1024]
    const int wave = (int)((blockIdx.x * blockDim.x + threadIdx.x) >> 5);
    const int lane = (int)(threadIdx.x & 31u);
    const int t    = wave >> 2;          // 0..1023
    const int btile= (wave & 3) << 4;    // 0,16,32,48
    const int m    = lane & 15;          // row within tile
    const int kh   = lane >> 4;          // 0: K=0..1, 1: K=2..3

    const float* rowp = enc + ((size_t)t * 64 + (size_t)(btile + m)) * 1024 + 2 * kh;
    const float* up   = u_e + 2 * kh;

    v8f c = {0.f, 0.f, 0.f, 0.f, 0.f, 0.f, 0.f, 0.f};

    for (int k = 0; k < 1024; k += 16) {
        __builtin_prefetch((const void*)(rowp + k + 256), 0, 1);
        v2f a0 = *(const v2f*)(rowp + k);
        v2f b0 = *(const v2f*)(up   + k);
        v2f a1 = *(const v2f*)(rowp + k + 4);
        v2f b1 = *(const v2f*)(up   + k + 4);
        v2f a2 = *(const v2f*)(rowp + k + 8);
        v2f b2 = *(const v2f*)(up   + k + 8);
        v2f a3 = *(const v2f*)(rowp + k + 12);
        v2f b3 = *(const v2f*)(up   + k + 12);
        c = __builtin_amdgcn_wmma_f32_16x16x4_f32(false, a0, false, b0, (short)0, c, false, false);
        c = __builtin_amdgcn_wmma_f32_16x16x4_f32(false, a1, false, b1, (short)0, c, false, false);
        c = __builtin_amdgcn_wmma_f32_16x16x4_f32(false, a2, false, b2, (short)0, c, false, false);
        c = __builtin_amdgcn_wmma_f32_16x16x4_f32(false, a3, false, b3, (short)0, c, false, false);
    }

    // Distance weight: log2(|t - time_step| + 2)
    const int ts = *time_step_p;
    int dd = t - ts; if (dd < 0) dd = -dd;
    const float w = log2f((float)(dd + 2));

    // Lane 0 holds rows 0..7 (N=0 column), lane 16 holds rows 8..15.
    if (lane == 0 || lane == 16) {
        const int mbase = (lane >> 4) * 8;
#pragma unroll
        for (int j = 0; j < 8; ++j) {
            const int b = btile + mbase + j;
            float s = (c[j] + score_h[b]) / w;
            if (mask[(size_t)b * 1024 + t] == 0) s = NEG_INF_F;
            scores[(size_t)b * 1024 + t] = s;
        }
    }
}

// ---------------------------------------------------------------------------
// Kernel 4: softmax over T per batch row; out is [B, 1, T] flat.
// ---------------------------------------------------------------------------
__global__ void k_softmax(const float* __restrict__ scores,
                          float* __restrict__ out) {
    __shared__ float red[256];
    int b = blockIdx.x;
    const float* row = scores + (size_t)b * 1024;

    float mx = -INFINITY;
    for (int t = threadIdx.x; t < 1024; t += 256) mx = fmaxf(mx, row[t]);
    red[threadIdx.x] = mx; __syncthreads();
    for (int s = 128; s > 0; s >>= 1) {
        if (threadIdx.x < (unsigned)s)
            red[threadIdx.x] = fmaxf(red[threadIdx.x], red[threadIdx.x + s]);
        __syncthreads();
    }
    mx = red[0]; __syncthreads();

    float sum = 0.f;
    for (int t = threadIdx.x; t < 1024; t += 256) sum += expf(row[t] - mx);
    red[threadIdx.x] = sum; __syncthreads();
    for (int s = 128; s > 0; s >>= 1) {
        if (threadIdx.x < (unsigned)s) red[threadIdx.x] += red[threadIdx.x + s];
        __syncthreads();
    }
    const float inv = 1.f / red[0];

    for (int t = threadIdx.x; t < 1024; t += 256)
        out[(size_t)b * 1024 + t] = expf(row[t] - mx) * inv;
}

// ---------------------------------------------------------------------------
extern "C" void kernel_launch(void* const* d_in, const int* in_sizes, int n_in,
                              void* d_out, int out_size, void* d_ws, size_t ws_size,
                              hipStream_t stream) {
    (void)in_sizes; (void)n_in; (void)out_size; (void)ws_size;

    const float* av   = (const float*)d_in[0];  // [64, 1024]
    const float* enc  = (const float*)d_in[1];  // [1024, 64, 1024]
    const float* W    = (const float*)d_in[2];  // [1024, 2048]
    const float* ba   = (const float*)d_in[3];  // [1024]
    const float* v    = (const float*)d_in[4];  // [1024]
    const int*   mask = (const int*)d_in[5];    // [64, 1024]
    const int*   ts   = (const int*)d_in[6];    // scalar time_step
    // d_in[7] = max_len (shapes hardcoded to 1024)

    float* ws      = (float*)d_ws;
    float* u       = ws;            // 2048 floats (u_h | u_e)
    float* c0      = ws + 2048;     // 1 float
    float* score_h = ws + 2064;     // 64 floats
    float* scores  = ws + 4096;     // 64*1024 floats

    float* out = (float*)d_out;     // [64, 1, 1024]

    k_project_v<<<8, 256, 0, stream>>>(W, v, ba, u, c0);
    k_score_h <<<64, 256, 0, stream>>>(av, u, c0, score_h);
    // 4096 tiles (1024 t * 4 b-tiles), 8 waves per 256-thread block
    k_score_wmma<<<512, 256, 0, stream>>>(enc, u + 1024, score_h, mask, ts, scores);
    k_softmax <<<64, 256, 0, stream>>>(scores, out);
}